// G2_GNN_42769284334146
// MI455X (gfx1250) — compile-verified
//
#include <hip/hip_runtime.h>
#include <math.h>

// G2-GNN forward for MI455X (gfx1250, wave32).
// GEMMs use V_WMMA_F32_16X16X4_F32 (fp32-exact vs reference).
// Edge aggregation uses float4 gathers + global f32 atomics; degrees computed once.

typedef __attribute__((ext_vector_type(2))) float v2f;
typedef __attribute__((ext_vector_type(8))) float v8f;

#define NFEAT 128
#define NHID  64
#define NCLASS 40

static __device__ __forceinline__ v8f wmma4(v2f a, v2f b, v8f c) {
  // (neg_a, A, neg_b, B, c_mod, C, reuse_a, reuse_b)
  return __builtin_amdgcn_wmma_f32_16x16x4_f32(false, a, false, b, (short)0, c,
                                               false, false);
}

// ---------------------------------------------------------------- utilities
__global__ void fill_zero_kernel(float* __restrict__ p, long long n) {
  long long i = (long long)blockIdx.x * blockDim.x + threadIdx.x;
  if (i < n) p[i] = 0.0f;
}

__global__ void deg_kernel(const int* __restrict__ src, const int* __restrict__ dst,
                           float* __restrict__ degS, float* __restrict__ degD, int E) {
  int e = blockIdx.x * blockDim.x + threadIdx.x;
  if (e >= E) return;
  atomicAdd(&degS[src[e]], 1.0f);
  atomicAdd(&degD[dst[e]], 1.0f);
}

// ------------------------------------------------------------- encoder GEMM
// X = relu(x @ Wenc + benc);  x: M x 128, Wenc: 128 x 64
__global__ void enc_kernel(const float* __restrict__ x, const float* __restrict__ W,
                           const float* __restrict__ bias, float* __restrict__ out,
                           int M) {
  const int lane = threadIdx.x & 31;
  const int wave = threadIdx.x >> 5;
  const int m0 = (blockIdx.x * 8 + wave) * 16;
  if (m0 >= M) return;                       // wave-uniform: EXEC stays all-ones
  const int mr = lane & 15;
  const int kh = (lane >> 4) << 1;
  v8f acc[4] = {};
  const float* arow = x + (long long)(m0 + mr) * NFEAT;
  for (int k0 = 0; k0 < NFEAT; k0 += 4) {
    v2f a = *(const v2f*)(arow + k0 + kh);
#pragma unroll
    for (int nt = 0; nt < 4; ++nt) {
      v2f b;
      b.x = W[(k0 + kh) * NHID + nt * 16 + mr];
      b.y = W[(k0 + kh + 1) * NHID + nt * 16 + mr];
      acc[nt] = wmma4(a, b, acc[nt]);
    }
  }
  const int rbase = (lane >> 4) << 3;
#pragma unroll
  for (int nt = 0; nt < 4; ++nt) {
    const int n = nt * 16 + mr;
    const float bn = bias[n];
#pragma unroll
    for (int r = 0; r < 8; ++r) {
      float v = acc[nt][r] + bn;
      out[(long long)(m0 + rbase + r) * NHID + n] = fmaxf(v, 0.0f);
    }
  }
}

// -------------------------------------------------- mean-aggregation scatter
// agg[dst] += X[src]   (divide by degree later, inside the GEMM kernel)
__global__ void scatter_kernel(const float* __restrict__ X, const int* __restrict__ src,
                               const int* __restrict__ dst, float* __restrict__ agg,
                               int E) {
  long long t = (long long)blockIdx.x * blockDim.x + threadIdx.x;
  int e = (int)(t >> 4);
  if (e >= E) return;
  int f = (int)(t & 15) * 4;
  int s = src[e], d = dst[e];
  const float4 v = *(const float4*)(X + (long long)s * NHID + f);
  float* p = agg + (long long)d * NHID + f;
  atomicAdd(p + 0, v.x);
  atomicAdd(p + 1, v.y);
  atomicAdd(p + 2, v.z);
  atomicAdd(p + 3, v.w);
}

// ------------------------------------------- fused dual SAGE GEMM (WMMA f32)
// mean = agg / max(degD,1)
// Xn = relu(mean@Wl  + bl  + X@Wr )
// Xg = relu(mean@Wlg + blg + X@Wrg)
__global__ void sage_kernel(const float* __restrict__ X, const float* __restrict__ agg,
                            const float* __restrict__ degD,
                            const float* __restrict__ Wl, const float* __restrict__ bl,
                            const float* __restrict__ Wr,
                            const float* __restrict__ Wlg, const float* __restrict__ blg,
                            const float* __restrict__ Wrg,
                            float* __restrict__ Xn, float* __restrict__ Xg, int M) {
  const int lane = threadIdx.x & 31;
  const int wave = threadIdx.x >> 5;
  const int m0 = (blockIdx.x * 8 + wave) * 16;
  if (m0 >= M) return;
  const int mr = lane & 15;
  const int kh = (lane >> 4) << 1;
  const long long row = (long long)(m0 + mr) * NHID;
  const float dinv = 1.0f / fmaxf(degD[m0 + mr], 1.0f);
  v8f aN[4] = {};
  v8f aG[4] = {};
  for (int k0 = 0; k0 < NHID; k0 += 4) {
    v2f am = *(const v2f*)(agg + row + k0 + kh);
    am.x *= dinv;
    am.y *= dinv;
    v2f ax = *(const v2f*)(X + row + k0 + kh);
    const int r0 = (k0 + kh) * NHID;
    const int r1 = (k0 + kh + 1) * NHID;
#pragma unroll
    for (int nt = 0; nt < 4; ++nt) {
      const int c = nt * 16 + mr;
      v2f b;
      b.x = Wl[r0 + c];  b.y = Wl[r1 + c];
      aN[nt] = wmma4(am, b, aN[nt]);
      b.x = Wr[r0 + c];  b.y = Wr[r1 + c];
      aN[nt] = wmma4(ax, b, aN[nt]);
      b.x = Wlg[r0 + c]; b.y = Wlg[r1 + c];
      aG[nt] = wmma4(am, b, aG[nt]);
      b.x = Wrg[r0 + c]; b.y = Wrg[r1 + c];
      aG[nt] = wmma4(ax, b, aG[nt]);
    }
  }
  const int rbase = (lane >> 4) << 3;
#pragma unroll
  for (int nt = 0; nt < 4; ++nt) {
    const int n = nt * 16 + mr;
    const float bn = bl[n];
    const float bg = blg[n];
#pragma unroll
    for (int r = 0; r < 8; ++r) {
      const long long o = (long long)(m0 + rbase + r) * NHID + n;
      Xn[o] = fmaxf(aN[nt][r] + bn, 0.0f);
      Xg[o] = fmaxf(aG[nt][r] + bg, 0.0f);
    }
  }
}

// --------------------------------------------------------- G2 gate scatter
// tau_sum[src] += (Xg[src]-Xg[dst])^2   (P = 2.0)
__global__ void gate_kernel(const float* __restrict__ Xg, const int* __restrict__ src,
                            const int* __restrict__ dst, float* __restrict__ tau,
                            int E) {
  long long t = (long long)blockIdx.x * blockDim.x + threadIdx.x;
  int e = (int)(t >> 4);
  if (e >= E) return;
  int f = (int)(t & 15) * 4;
  int s = src[e], d = dst[e];
  const float4 a = *(const float4*)(Xg + (long long)s * NHID + f);
  const float4 b = *(const float4*)(Xg + (long long)d * NHID + f);
  float dx = a.x - b.x, dy = a.y - b.y, dz = a.z - b.z, dw = a.w - b.w;
  float* p = tau + (long long)s * NHID + f;
  atomicAdd(p + 0, dx * dx);
  atomicAdd(p + 1, dy * dy);
  atomicAdd(p + 2, dz * dz);
  atomicAdd(p + 3, dw * dw);
}

// --------------------------------------------------------- gated update
// tau = tanh(tau_sum / max(degS,1));  X = (1-tau)*X + tau*Xn
__global__ void update_kernel(float* __restrict__ X, const float* __restrict__ Xn,
                              const float* __restrict__ tau,
                              const float* __restrict__ degS, long long total) {
  long long i = (long long)blockIdx.x * blockDim.x + threadIdx.x;
  if (i >= total) return;
  int node = (int)(i >> 6);  // NHID == 64
  float t = tanhf(tau[i] / fmaxf(degS[node], 1.0f));
  X[i] = (1.0f - t) * X[i] + t * Xn[i];
}

// ------------------------------------------------------------- decoder GEMM
// out = X @ Wdec + bdec;  Wdec: 64 x 40  (3 n-tiles, column-masked)
__global__ void dec_kernel(const float* __restrict__ X, const float* __restrict__ W,
                           const float* __restrict__ bias, float* __restrict__ out,
                           int M) {
  const int lane = threadIdx.x & 31;
  const int wave = threadIdx.x >> 5;
  const int m0 = (blockIdx.x * 8 + wave) * 16;
  if (m0 >= M) return;
  const int mr = lane & 15;
  const int kh = (lane >> 4) << 1;
  v8f acc[3] = {};
  const float* arow = X + (long long)(m0 + mr) * NHID;
  for (int k0 = 0; k0 < NHID; k0 += 4) {
    v2f a = *(const v2f*)(arow + k0 + kh);
#pragma unroll
    for (int nt = 0; nt < 3; ++nt) {
      const int n = nt * 16 + mr;
      v2f b = {0.0f, 0.0f};
      if (n < NCLASS) {
        b.x = W[(k0 + kh) * NCLASS + n];
        b.y = W[(k0 + kh + 1) * NCLASS + n];
      }
      acc[nt] = wmma4(a, b, acc[nt]);
    }
  }
  const int rbase = (lane >> 4) << 3;
#pragma unroll
  for (int nt = 0; nt < 3; ++nt) {
    const int n = nt * 16 + mr;
    if (n < NCLASS) {
      const float bn = bias[n];
#pragma unroll
      for (int r = 0; r < 8; ++r)
        out[(long long)(m0 + rbase + r) * NCLASS + n] = acc[nt][r] + bn;
    }
  }
}

// ------------------------------------------------------------------ launch
extern "C" void kernel_launch(void* const* d_in, const int* in_sizes, int n_in,
                              void* d_out, int out_size, void* d_ws, size_t ws_size,
                              hipStream_t stream) {
  const float* x    = (const float*)d_in[0];
  const int*   ei   = (const int*)d_in[1];
  const float* Wenc = (const float*)d_in[2];
  const float* benc = (const float*)d_in[3];
  const float* Wl   = (const float*)d_in[4];
  const float* bl   = (const float*)d_in[5];
  const float* Wr   = (const float*)d_in[6];
  const float* Wlg  = (const float*)d_in[7];
  const float* blg  = (const float*)d_in[8];
  const float* Wrg  = (const float*)d_in[9];
  const float* Wdec = (const float*)d_in[10];
  const float* bdec = (const float*)d_in[11];

  const int Nn = in_sizes[0] / NFEAT;   // 50000 (multiple of 16)
  const int E  = in_sizes[1] / 2;       // 1.6M
  const int* src = ei;
  const int* dst = ei + E;

  float* ws = (float*)d_ws;
  const long long nf = (long long)Nn * NHID;
  float* X    = ws;
  float* Xn   = ws + nf;
  float* Xg   = ws + 2 * nf;
  float* agg  = ws + 3 * nf;
  float* tau  = ws + 4 * nf;
  float* degD = ws + 5 * nf;
  float* degS = degD + Nn;

  const dim3 blk(256);
  const int gemm_blocks = (Nn + 127) / 128;                // 8 waves x 16 rows
  const int edge_blocks = (E + 255) / 256;
  const long long sth = (long long)E * 16;                 // 16 threads / edge
  const int scat_blocks = (int)((sth + 255) / 256);
  const int nf_blocks = (int)((nf + 255) / 256);

  // degrees (loop-invariant: compute once)
  fill_zero_kernel<<<(2 * Nn + 255) / 256, blk, 0, stream>>>(degD, 2LL * Nn);
  deg_kernel<<<edge_blocks, blk, 0, stream>>>(src, dst, degS, degD, E);

  // encoder
  enc_kernel<<<gemm_blocks, blk, 0, stream>>>(x, Wenc, benc, X, Nn);

  for (int layer = 0; layer < 4; ++layer) {
    fill_zero_kernel<<<nf_blocks, blk, 0, stream>>>(agg, nf);
    scatter_kernel<<<scat_blocks, blk, 0, stream>>>(X, src, dst, agg, E);
    sage_kernel<<<gemm_blocks, blk, 0, stream>>>(X, agg, degD, Wl, bl, Wr,
                                                 Wlg, blg, Wrg, Xn, Xg, Nn);
    fill_zero_kernel<<<nf_blocks, blk, 0, stream>>>(tau, nf);
    gate_kernel<<<scat_blocks, blk, 0, stream>>>(Xg, src, dst, tau, E);
    update_kernel<<<nf_blocks, blk, 0, stream>>>(X, Xn, tau, degS, nf);
  }

  dec_kernel<<<gemm_blocks, blk, 0, stream>>>(X, Wdec, bdec, (float*)d_out, Nn);
}